// HomographyLoss_45432164057681
// MI455X (gfx1250) — compile-verified
//
#include <hip/hip_runtime.h>

typedef float v2f __attribute__((ext_vector_type(2)));
typedef float v8f __attribute__((ext_vector_type(8)));
typedef unsigned int u32x4 __attribute__((ext_vector_type(4)));
typedef int i32x4 __attribute__((ext_vector_type(4)));
typedef int i32x8 __attribute__((ext_vector_type(8)));

#define IMG_H 512
#define IMG_W 512
#define TILE 16
#define NTILES 98304                     // 96 planes * 32 * 32 tiles
#define WAVES_PER_BLOCK 2
#define BLOCK_THREADS 64
#define NBLOCKS (NTILES / WAVES_PER_BLOCK)   // 49152

#define HALO 5
#define IN_ROWS 26                       // 16 + 2*5
#define IN_COLS 28                       // 26 + 2 pad -> K tiles to 7 chunks of 4
#define PLANE_ELEMS (IN_ROWS * IN_COLS)  // 728
#define HB_ELEMS (28 * 16)               // 448

// banded Gaussian weight: g[j] for j in [0,10], else 0
__device__ __forceinline__ float bandw(const float* sg, int j) {
    int jc = j < 0 ? 0 : (j > 10 ? 10 : j);
    float v = sg[jc];
    return (j == jc) ? v : 0.0f;
}

__global__ __launch_bounds__(BLOCK_THREADS) void ssim_l1_tiles(
    const float* __restrict__ warped, const float* __restrict__ target,
    float* __restrict__ partL1, float* __restrict__ partSSIM)
{
    __shared__ float sP[WAVES_PER_BLOCK][2][PLANE_ELEMS];  // 0 = x, 1 = y
    __shared__ float sH[WAVES_PER_BLOCK][5][HB_ELEMS];     // H-pass scratch (5 quantities)
    __shared__ float sg[11];
    __shared__ float red1[BLOCK_THREADS];
    __shared__ float red2[BLOCK_THREADS];

    const int tid = threadIdx.x;
    const int wv  = tid >> 5;     // wave in block
    const int ln  = tid & 31;     // lane
    const int hi  = ln >> 4;      // lane-half select (K/M split)
    const int lo  = ln & 15;      // M (A) / N (B,D) index

    // Normalized 1D Gaussian (sigma=1.5, K=11)
    if (tid < 11) {
        float sum = 0.0f;
        for (int i = 0; i < 11; ++i) {
            float d = (float)(i - 5);
            sum += expf(-(d * d) / 4.5f);
        }
        float d = (float)(tid - 5);
        sg[tid] = expf(-(d * d) / 4.5f) / sum;
    }

    // ---- tile coordinates ----
    const int t  = blockIdx.x * WAVES_PER_BLOCK + wv;
    const int tx = t & 31;
    const int ty = (t >> 5) & 31;
    const int pl = t >> 10;                  // plane 0..95
    const int r0 = ty * TILE;
    const int c0 = tx * TILE;
    const float* wp = warped + (size_t)pl * (IMG_H * IMG_W);
    const float* tp = target + (size_t)pl * (IMG_H * IMG_W);

    float* pw = &sP[wv][0][0];
    float* pt = &sP[wv][1][0];
    float* hb = &sH[wv][0][0];

    const bool interior = (tx >= 1) && (tx <= 30) && (ty >= 1) && (ty <= 30);

    if (interior) {
        // ---- TDM: one 2D tensor DMA per image plane (26 rows x 28 cols) ----
        const unsigned long long gaW =
            (unsigned long long)(uintptr_t)(wp + ((r0 - HALO) * IMG_W + (c0 - HALO)));
        const unsigned long long gaT =
            (unsigned long long)(uintptr_t)(tp + ((r0 - HALO) * IMG_W + (c0 - HALO)));

        // D# group1: data_size=4B (2), tensor_dim0=512, tensor_dim1=512,
        //            tile_dim0=28, tile_dim1=26, tensor_dim0_stride=512
        i32x8 g1 = { 0x00020000,            // [17:16] data_size = 2 (4 bytes)
                     (int)(512u << 16),     // tensor_dim0[15:0] at bits [31:16]
                     (int)(512u << 16),     // tensor_dim0 hi=0, tensor_dim1[15:0] at [31:16]
                     (int)(28u  << 16),     // tensor_dim1 hi=0, tile_dim0=28 at [31:16]
                     26,                    // tile_dim1=26, tile_dim2=0
                     512,                   // tensor_dim0_stride low 32
                     0, 0 };
        i32x4 gz  = { 0, 0, 0, 0 };
        i32x8 gz8 = { 0, 0, 0, 0, 0, 0, 0, 0 };

        u32x4 g0w = { 1u,                                  // count=1
                      (unsigned)(uintptr_t)pw,             // lds_addr
                      (unsigned)gaW,                       // global_addr[31:0]
                      ((unsigned)((gaW >> 32) & 0x01FFFFFFull)) | 0x80000000u }; // addr[56:32] | type=2
        u32x4 g0t = { 1u,
                      (unsigned)(uintptr_t)pt,
                      (unsigned)gaT,
                      ((unsigned)((gaT >> 32) & 0x01FFFFFFull)) | 0x80000000u };

        __builtin_amdgcn_tensor_load_to_lds(g0w, g1, gz, gz, gz8, 0);
        __builtin_amdgcn_tensor_load_to_lds(g0t, g1, gz, gz, gz8, 0);
    } else {
        // ---- boundary: per-lane async global->LDS with zero padding ----
        const unsigned ldsW0 = (unsigned)(uintptr_t)pw;
        const unsigned ldsT0 = (unsigned)(uintptr_t)pt;
        const unsigned long long baseW = (unsigned long long)(uintptr_t)wp;
        const unsigned long long baseT = (unsigned long long)(uintptr_t)tp;
        for (int i = ln; i < PLANE_ELEMS; i += 32) {
            int row = i / IN_COLS;
            int col = i - row * IN_COLS;
            int gr = r0 - HALO + row;
            int gc = c0 - HALO + col;
            bool ok = (col < 26) && (gr >= 0) && (gr < IMG_H) && (gc >= 0) && (gc < IMG_W);
            if (ok) {
                int goff = (gr * IMG_W + gc) * 4;
                unsigned la = ldsW0 + (unsigned)i * 4u;
                unsigned lb = ldsT0 + (unsigned)i * 4u;
                asm volatile("global_load_async_to_lds_b32 %0, %1, %2"
                             :: "v"(la), "v"(goff), "s"(baseW) : "memory");
                asm volatile("global_load_async_to_lds_b32 %0, %1, %2"
                             :: "v"(lb), "v"(goff), "s"(baseT) : "memory");
            } else {
                pw[i] = 0.0f;
                pt[i] = 0.0f;
            }
        }
    }
    __syncthreads();   // gaussian table visible (DMA may still be in flight)

    // ---- hoisted banded weights (shared by H-pass B and V-pass A) ----
    v2f wgt[7];
#pragma unroll
    for (int ck = 0; ck < 7; ++ck) {
        int kb = ck * 4 + 2 * hi;
        wgt[ck].x = bandw(sg, kb - lo);
        wgt[ck].y = bandw(sg, kb + 1 - lo);
    }

    // ---- wait for tile data (TDM or async path) ----
    __builtin_amdgcn_s_wait_tensorcnt(0);
    asm volatile("s_wait_asynccnt 0x0" ::: "memory");

    // ---- horizontal pass: 5 quantities share the banded-Gaussian B operand ----
#pragma unroll
    for (int rg = 0; rg < 2; ++rg) {
        v8f acc[5] = {};
        const int arow  = rg * 16 + lo;
        const bool rok  = arow < IN_ROWS;
        const int rbase = arow * IN_COLS;
#pragma unroll
        for (int ck = 0; ck < 7; ++ck) {
            const int kb = ck * 4 + 2 * hi;
            v2f xv, yv;
            if (rok) {
                xv.x = pw[rbase + kb];
                xv.y = pw[rbase + kb + 1];
                yv.x = pt[rbase + kb];
                yv.y = pt[rbase + kb + 1];
            } else {
                xv.x = 0.0f; xv.y = 0.0f;
                yv.x = 0.0f; yv.y = 0.0f;
            }
            v2f a[5];
            a[0] = xv;
            a[1] = yv;
            a[2] = xv * xv;
            a[3] = yv * yv;
            a[4] = xv * yv;
#pragma unroll
            for (int p = 0; p < 5; ++p) {
                acc[p] = __builtin_amdgcn_wmma_f32_16x16x4_f32(
                            false, a[p], false, wgt[ck], (short)0, acc[p], false, false);
            }
        }
        // D layout: element v -> row v + 8*hi, col lo
#pragma unroll
        for (int p = 0; p < 5; ++p) {
#pragma unroll
            for (int v = 0; v < 8; ++v) {
                int orow = rg * 16 + v + 8 * hi;
                if (rg == 0 || orow < 28)
                    hb[p * HB_ELEMS + orow * 16 + lo] = acc[p][v];
            }
        }
    }
    __syncthreads();

    // ---- vertical pass: out = Wv(banded gaussian) * out_h ----
    v8f res[5] = {};
#pragma unroll
    for (int ck = 0; ck < 7; ++ck) {
        const int kb = ck * 4 + 2 * hi;
#pragma unroll
        for (int p = 0; p < 5; ++p) {
            v2f b;
            b.x = hb[p * HB_ELEMS + kb * 16 + lo];
            b.y = hb[p * HB_ELEMS + (kb + 1) * 16 + lo];
            res[p] = __builtin_amdgcn_wmma_f32_16x16x4_f32(
                        false, wgt[ck], false, b, (short)0, res[p], false, false);
        }
    }

    // ---- SSIM map + L1 partials ----
    const float C1 = 1.0e-4f;   // 0.01^2
    const float C2 = 9.0e-4f;   // 0.03^2
    float l1acc = 0.0f, ssacc = 0.0f;
#pragma unroll
    for (int v = 0; v < 8; ++v) {
        float mu1 = res[0][v], mu2 = res[1][v];
        float mu1s = mu1 * mu1, mu2s = mu2 * mu2, mu12 = mu1 * mu2;
        float s1  = res[2][v] - mu1s;
        float s2  = res[3][v] - mu2s;
        float s12 = res[4][v] - mu12;
        float num = (2.0f * mu12 + C1) * (2.0f * s12 + C2);
        float den = (mu1s + mu2s + C1) * (s1 + s2 + C2);
        ssacc += num / den;

        int prow = v + 8 * hi;
        int pcol = lo;
        float a = pw[(prow + HALO) * IN_COLS + (pcol + HALO)];
        float b = pt[(prow + HALO) * IN_COLS + (pcol + HALO)];
        l1acc += fabsf(a - b);
    }

    // ---- deterministic block reduction ----
    red1[tid] = l1acc;
    red2[tid] = ssacc;
    __syncthreads();
    for (int s = BLOCK_THREADS / 2; s > 0; s >>= 1) {
        if (tid < s) {
            red1[tid] += red1[tid + s];
            red2[tid] += red2[tid + s];
        }
        __syncthreads();
    }
    if (tid == 0) {
        partL1[blockIdx.x]   = red1[0];
        partSSIM[blockIdx.x] = red2[0];
    }
}

__global__ __launch_bounds__(256) void ssim_l1_final(
    const float* __restrict__ partL1, const float* __restrict__ partSSIM,
    int n, float* __restrict__ out)
{
    __shared__ float r1[256];
    __shared__ float r2[256];
    int tid = threadIdx.x;
    float a = 0.0f, b = 0.0f;
    for (int i = tid; i < n; i += 256) {
        a += partL1[i];
        b += partSSIM[i];
    }
    r1[tid] = a; r2[tid] = b;
    __syncthreads();
    for (int s = 128; s > 0; s >>= 1) {
        if (tid < s) { r1[tid] += r1[tid + s]; r2[tid] += r2[tid + s]; }
        __syncthreads();
    }
    if (tid == 0) {
        const float N = 25165824.0f;  // 32*3*512*512
        float l1 = r1[0] / N;
        float ss = r2[0] / N;
        out[0] = 0.5f * l1 + 0.5f * (1.0f - ss);
    }
}

extern "C" void kernel_launch(void* const* d_in, const int* in_sizes, int n_in,
                              void* d_out, int out_size, void* d_ws, size_t ws_size,
                              hipStream_t stream) {
    (void)in_sizes; (void)n_in; (void)out_size; (void)ws_size;
    const float* warped = (const float*)d_in[0];
    const float* target = (const float*)d_in[1];
    float* out = (float*)d_out;
    float* ws  = (float*)d_ws;
    float* partL1   = ws;
    float* partSSIM = ws + NBLOCKS;

    ssim_l1_tiles<<<NBLOCKS, BLOCK_THREADS, 0, stream>>>(warped, target, partL1, partSSIM);
    ssim_l1_final<<<1, 256, 0, stream>>>(partL1, partSSIM, NBLOCKS, out);
}